// SpGraphAttentionLayer_11364483465752
// MI455X (gfx1250) — compile-verified
//
#include <hip/hip_runtime.h>
#include <hip/hip_bf16.h>
#include <math.h>

// ---------------------------------------------------------------------------
// GAT layer for MI455X (gfx1250, wave32).
//   B=8, N=20000, F_IN=F_OUT=128, E=320000
// Phases:
//   1) h = x @ W            -- WMMA f32 16x16x4 GEMM, W packed in LDS as
//                              K-groups-of-4 so each B fragment is one
//                              bank-conflict-free ds_load_b64; A fragments
//                              register-prefetched one K-step ahead.
//   2) hd1 = h@a1, hd2=h@a2 -- one wave per row, shuffle reduction
//   3) per-edge weight + atomic scatter into d_out (seg) and rowsum
//   4) finalize: elu(seg / rowsum) in place in d_out
// ---------------------------------------------------------------------------

typedef __attribute__((ext_vector_type(2))) float v2f;
typedef __attribute__((ext_vector_type(8))) float v8f;

#define GAT_F   128
#define GAT_B   8
#define GAT_ALPHA 0.2f

// ---------------------------------------------------------------------------
// Kernel 1: h[row,:] = x[row,:] @ W,  rows = B*N (multiple of 128).
// Block = 256 threads (8 waves). Each block: 128 rows; each wave: 16 rows x 128 cols.
// ---------------------------------------------------------------------------
__global__ void __launch_bounds__(256)
gat_gemm_wmma(const float* __restrict__ x, const float* __restrict__ W,
              float* __restrict__ h, int total_rows)
{
    // W packed as K-groups of 4: Ws[(g*128 + col)*4 + e] = W[4g+e][col].
    // B fragment for (g, half, col) = Ws[(g*128+col)*4 + half*2 + {0,1}]
    //   -> one 8B-aligned ds_load_b64; dword banks 4*l16+{0,1} (half 0) vs
    //      4*l16+{2,3} (half 1): all 64 banks, conflict-free. 64 KB total.
    __shared__ float Ws[GAT_F * GAT_F];

    const int tid = threadIdx.x;
    for (int i = tid; i < GAT_F * GAT_F; i += 256) {
        const int k   = i >> 7;          // row of W (K index)
        const int col = i & 127;         // col of W
        Ws[(((k >> 2) * GAT_F + col) << 2) + (k & 3)] = W[i];
    }
    __syncthreads();

    const int wave = tid >> 5;
    const int lane = tid & 31;
    const int half = lane >> 4;          // 0: lanes 0-15, 1: lanes 16-31
    const int l16  = lane & 15;

    const long rowBase = (long)blockIdx.x * 128 + wave * 16;
    if (rowBase >= total_rows) return;

    v8f acc[8] = {};                     // 8 col-tiles of 16 -> 128 output cols

    // A fragment source: lane (half,l16) reads x[rowBase+l16][g*4 + half*2 +{0,1}]
    const float* alane = x + (rowBase + l16) * GAT_F + half * 2;

    v2f a_cur, a_nxt;
    a_cur.x = alane[0];
    a_cur.y = alane[1];

    #pragma unroll 2
    for (int g = 0; g < GAT_F / 4; ++g) {
        if (g < GAT_F / 4 - 1) {         // prefetch next K-step's A fragment
            a_nxt.x = alane[(g + 1) * 4];
            a_nxt.y = alane[(g + 1) * 4 + 1];
        }
        #pragma unroll
        for (int ct = 0; ct < 8; ++ct) {
            const v2f bfrag =
                *(const v2f*)&Ws[(((g * GAT_F) + ct * 16 + l16) << 2) + half * 2];
            acc[ct] = __builtin_amdgcn_wmma_f32_16x16x4_f32(
                false, a_cur, false, bfrag, (short)0, acc[ct], false, false);
        }
        a_cur = a_nxt;
    }

    // C/D layout: VGPR i -> row i (lanes 0-15) / row i+8 (lanes 16-31), col = lane%16
    float* hrow = h + rowBase * GAT_F;
    #pragma unroll
    for (int ct = 0; ct < 8; ++ct) {
        #pragma unroll
        for (int i = 0; i < 8; ++i) {
            const int r = i + half * 8;
            hrow[(long)r * GAT_F + ct * 16 + l16] = acc[ct][i];
        }
    }
}

// ---------------------------------------------------------------------------
// Kernel 2: per-row dots hd1 = h.a1, hd2 = h.a2. One wave per row.
// ---------------------------------------------------------------------------
__global__ void __launch_bounds__(256)
gat_rowdot(const float* __restrict__ h, const float* __restrict__ a,
           float* __restrict__ hd1, float* __restrict__ hd2, int total_rows)
{
    const int wid  = (int)((blockIdx.x * (long)blockDim.x + threadIdx.x) >> 5);
    const int lane = threadIdx.x & 31;
    if (wid >= total_rows) return;

    const float4 hv = ((const float4*)(h + (long)wid * GAT_F))[lane];
    const float4 v1 = ((const float4*)a)[lane];             // a1 = a[0:128]
    const float4 v2 = ((const float4*)(a + GAT_F))[lane];   // a2 = a[128:256]

    float d1 = hv.x * v1.x + hv.y * v1.y + hv.z * v1.z + hv.w * v1.w;
    float d2 = hv.x * v2.x + hv.y * v2.y + hv.z * v2.z + hv.w * v2.w;

    #pragma unroll
    for (int off = 16; off > 0; off >>= 1) {
        d1 += __shfl_down(d1, off, 32);
        d2 += __shfl_down(d2, off, 32);
    }
    if (lane == 0) { hd1[wid] = d1; hd2[wid] = d2; }
}

// ---------------------------------------------------------------------------
// Kernel 3: per (edge, batch): weight + scatter. One wave per (e,b).
//   s = hd1[b,src] + hd2[b,dst];  w = exp(-leakyrelu(s))
//   outacc[b,src,:] += w * h[b,dst,:];  rowsum[b,src] += w
// ---------------------------------------------------------------------------
__global__ void __launch_bounds__(256)
gat_edge(const float* __restrict__ h, const float* __restrict__ hd1,
         const float* __restrict__ hd2, const int* __restrict__ edge,
         float* __restrict__ outacc, float* __restrict__ rowsum,
         int E, int N)
{
    const long wid  = (blockIdx.x * (long)blockDim.x + threadIdx.x) >> 5;
    const int  lane = threadIdx.x & 31;
    const long total = (long)E * GAT_B;
    if (wid >= total) return;

    const int e = (int)(wid >> 3);   // /B
    const int b = (int)(wid & 7);    // %B

    const int src = edge[e];
    const int dst = edge[E + e];

    const float s  = hd1[(long)b * N + src] + hd2[(long)b * N + dst];
    const float sc = s > 0.0f ? s : GAT_ALPHA * s;
    const float w  = expf(-sc);

    const float4 hv = ((const float4*)(h + ((long)b * N + dst) * GAT_F))[lane];

    float* outp = outacc + ((long)b * N + src) * GAT_F + lane * 4;
    atomicAdd(outp + 0, w * hv.x);
    atomicAdd(outp + 1, w * hv.y);
    atomicAdd(outp + 2, w * hv.z);
    atomicAdd(outp + 3, w * hv.w);
    if (lane == 0) atomicAdd(&rowsum[(long)b * N + src], w);
}

// ---------------------------------------------------------------------------
// Kernel 4: out = elu(out / rowsum), in place. One wave per row.
// ---------------------------------------------------------------------------
__global__ void __launch_bounds__(256)
gat_final(float* __restrict__ out, const float* __restrict__ rowsum, int total_rows)
{
    const int wid  = (int)((blockIdx.x * (long)blockDim.x + threadIdx.x) >> 5);
    const int lane = threadIdx.x & 31;
    if (wid >= total_rows) return;

    const float inv = 1.0f / rowsum[wid];
    float4* p = (float4*)(out + (long)wid * GAT_F) + lane;
    float4 v = *p;
    v.x *= inv; v.y *= inv; v.z *= inv; v.w *= inv;
    v.x = v.x > 0.0f ? v.x : expm1f(v.x);
    v.y = v.y > 0.0f ? v.y : expm1f(v.y);
    v.z = v.z > 0.0f ? v.z : expm1f(v.z);
    v.w = v.w > 0.0f ? v.w : expm1f(v.w);
    *p = v;
}

// ---------------------------------------------------------------------------
extern "C" void kernel_launch(void* const* d_in, const int* in_sizes, int n_in,
                              void* d_out, int out_size, void* d_ws, size_t ws_size,
                              hipStream_t stream)
{
    const float* x    = (const float*)d_in[0];   // (B, N, 128) f32
    const float* W    = (const float*)d_in[1];   // (128, 128)  f32
    const float* a    = (const float*)d_in[2];   // (256, 1)    f32
    const int*   edge = (const int*)d_in[3];     // (2, E)      i32

    const int E  = in_sizes[3] / 2;
    const int N  = (in_sizes[0] / GAT_B) / GAT_F;     // 20000
    const long BN = (long)GAT_B * N;                  // 160000 rows

    float* out = (float*)d_out;

    // workspace layout
    float* h      = (float*)d_ws;            // BN * 128
    float* hd1    = h + BN * GAT_F;          // BN
    float* hd2    = hd1 + BN;                // BN
    float* rowsum = hd2 + BN;                // BN

    // zero accumulators (d_out is the segment-sum accumulator)
    hipMemsetAsync(d_out, 0, sizeof(float) * BN * GAT_F, stream);
    hipMemsetAsync(rowsum, 0, sizeof(float) * BN, stream);

    // 1) h = x @ W   (128 rows per block, 8 waves * 16 rows)
    {
        const int blocks = (int)((BN + 127) / 128);
        gat_gemm_wmma<<<blocks, 256, 0, stream>>>(x, W, h, (int)BN);
    }
    // 2) row dots (one wave per row)
    {
        const long threads = BN * 32;
        gat_rowdot<<<(int)((threads + 255) / 256), 256, 0, stream>>>(h, a, hd1, hd2, (int)BN);
    }
    // 3) edge scatter (one wave per (edge, batch))
    {
        const long threads = (long)E * GAT_B * 32;
        gat_edge<<<(int)((threads + 255) / 256), 256, 0, stream>>>(h, hd1, hd2, edge,
                                                                   out, rowsum, E, N);
    }
    // 4) finalize
    {
        const long threads = BN * 32;
        gat_final<<<(int)((threads + 255) / 256), 256, 0, stream>>>(out, rowsum, (int)BN);
    }
}